// Net_15152644620565
// MI455X (gfx1250) — compile-verified
//
#include <hip/hip_runtime.h>
#include <hip/hip_bf16.h>

// ---------------------------------------------------------------------------
// Problem constants (from reference): NX=20, NC=5, NEXP=7, NCOL=121, B=128,
// L=41 (pad 48), C=11 (pad 16), DX=0.05, DC=0.2, BETA=50, scale=DX^2*DC^2=1e-4
// a : (128, 41, 41, 11, 11)  = 26,035,328 f32  (innermost: d, then c, y, x, b)
// ---------------------------------------------------------------------------

typedef float v2f __attribute__((ext_vector_type(2)));
typedef float v8f __attribute__((ext_vector_type(8)));

#define NEL_A 26035328L            // 128*41*41*121
#define NTILE 215168               // 128*41*41 color tiles
#define IMG   15488                // 128*121 spatial images

static __device__ __forceinline__ v8f wmma_f32(v2f a, v2f b, v8f c) {
  // D = A(16x4,f32) * B(4x16,f32) + C(16x16,f32)   -> v_wmma_f32_16x16x4_f32
  return __builtin_amdgcn_wmma_f32_16x16x4_f32(
      /*neg_a=*/false, a, /*neg_b=*/false, b,
      /*c_mod=*/(short)0, c, /*reuse_a=*/false, /*reuse_b=*/false);
}

// ---------------------------------------------------------------------------
// K0a: build padded Gaussian matrices from device scalars.
//   Mab: [0..2303]=Ma (48x48, sigma=alphag), [2304..4607]=Mb (sigma=betag)
//   Fab: [0..255]=Fa (16x16, sigma=alphaf), [256..511]=Fb row-reversed (betaf)
// ---------------------------------------------------------------------------
__global__ __launch_bounds__(256) void k_mats(float* __restrict__ Mab,
                                              float* __restrict__ Fab,
                                              const float* ag_p, const float* bg_p,
                                              const float* af_p, const float* bf_p) {
  int tid = threadIdx.x;
  float ag = *ag_p, bg = *bg_p, af = *af_p, bfv = *bf_p;
  float ia = 1.0f / (2.0f * ag * ag);
  float ib = 1.0f / (2.0f * bg * bg);
  for (int i = tid; i < 2304; i += 256) {
    int r = i / 48, c = i % 48;
    float va = 0.0f, vb = 0.0f;
    if (r < 41 && c < 41) {
      float d = (float)(r - c) * 0.05f;
      va = __expf(-d * d * ia);
      vb = __expf(-d * d * ib);
    }
    Mab[i] = va;
    Mab[2304 + i] = vb;
  }
  float ja = 1.0f / (2.0f * af * af);
  float jb = 1.0f / (2.0f * bfv * bfv);
  if (tid < 256) {
    int r = tid / 16, c = tid % 16;
    float va = 0.0f, vb = 0.0f;
    if (r < 11 && c < 11) {
      float d = (float)(r - c) * 0.2f;
      va = __expf(-d * d * ja);
      float df = (float)((10 - r) - c) * 0.2f;   // flip folded into matrix
      vb = __expf(-df * df * jb);
    }
    Fab[tid] = va;
    Fab[256 + tid] = vb;
  }
}

// K0b: a = 0.5 everywhere
__global__ __launch_bounds__(256) void k_init(float* __restrict__ p, long n) {
  long i = (long)blockIdx.x * 256 + threadIdx.x;
  if (i < n) p[i] = 0.5f;
}

// ---------------------------------------------------------------------------
// K1: spatial pass. One workgroup (4 waves) per (b, cd) image.
//   out = mug * Ma*A*Ma - nug * Mb*A*Mb     (Ma, Mb symmetric)
// ---------------------------------------------------------------------------
__global__ __launch_bounds__(128) void k_spatial(const float* __restrict__ in,
                                                 float* __restrict__ out,
                                                 const float* __restrict__ Mab,
                                                 const float* mug_p,
                                                 const float* nug_p) {
  __shared__ float Ms[4608];        // Ma | Mb, each 48x48
  __shared__ float As[2304];        // padded input image
  __shared__ float Ta[2304];        // Ma*A
  __shared__ float Tb[2304];        // Mb*A
  int tid = threadIdx.x;
  int b = blockIdx.x / 121;
  int cd = blockIdx.x % 121;

  for (int i = tid; i < 4608; i += 128) Ms[i] = Mab[i];

  const float* src = in + (size_t)b * (1681 * 121) + cd;
  for (int i = tid; i < 2304; i += 128) {
    int x = i / 48, y = i % 48;
    As[i] = (x < 41 && y < 41) ? src[(x * 41 + y) * 121] : 0.0f;
  }
  __syncthreads();

  int wave = tid >> 5, lane = tid & 31;
  int m = lane & 15;                 // A-frag row / B-frag col
  int kb = (lane >> 4) << 1;         // K sub-offset {0,2}
  int rhalf = (lane >> 4) << 3;      // C/D row offset {0,8}

  // ---- Step 1: Ta = Ma*A (tiles 0..8), Tb = Mb*A (tiles 9..17) ----
  for (int t = wave; t < 18; t += 4) {
    const float* M = (t < 9) ? Ms : (Ms + 2304);
    float* T = (t < 9) ? Ta : Tb;
    int tt = (t < 9) ? t : t - 9;
    int it = tt / 3, jt = tt % 3;
    v8f acc = {};
    for (int k4 = 0; k4 < 48; k4 += 4) {
      v2f af, bf;
      af.x = M[(it * 16 + m) * 48 + k4 + kb];
      af.y = M[(it * 16 + m) * 48 + k4 + kb + 1];
      bf.x = As[(k4 + kb) * 48 + jt * 16 + m];
      bf.y = As[(k4 + kb + 1) * 48 + jt * 16 + m];
      acc = wmma_f32(af, bf, acc);
    }
#pragma unroll
    for (int r = 0; r < 8; ++r)
      T[(it * 16 + r + rhalf) * 48 + jt * 16 + m] = acc[r];
  }
  __syncthreads();

  // ---- Step 2: R = mug*Ta*Ma - nug*Tb*Mb (symmetric -> B straight from M) ----
  float mug = *mug_p, nug = *nug_p;
  float* dst = out + (size_t)b * (1681 * 121) + cd;
  for (int t = wave; t < 9; t += 4) {
    int it = t / 3, jt = t % 3;
    v8f accA = {}, accB = {};
    for (int k4 = 0; k4 < 48; k4 += 4) {
      v2f af, bf;
      af.x = Ta[(it * 16 + m) * 48 + k4 + kb];
      af.y = Ta[(it * 16 + m) * 48 + k4 + kb + 1];
      bf.x = Ms[(k4 + kb) * 48 + jt * 16 + m];
      bf.y = Ms[(k4 + kb + 1) * 48 + jt * 16 + m];
      accA = wmma_f32(af, bf, accA);
      v2f ag, bg;
      ag.x = Tb[(it * 16 + m) * 48 + k4 + kb];
      ag.y = Tb[(it * 16 + m) * 48 + k4 + kb + 1];
      bg.x = Ms[2304 + (k4 + kb) * 48 + jt * 16 + m];
      bg.y = Ms[2304 + (k4 + kb + 1) * 48 + jt * 16 + m];
      accB = wmma_f32(ag, bg, accB);
    }
#pragma unroll
    for (int r = 0; r < 8; ++r) {
      int x = it * 16 + r + rhalf;
      int y = jt * 16 + m;
      if (x < 41 && y < 41)
        dst[(x * 41 + y) * 121] = mug * accA[r] - nug * accB[r];
    }
  }
}

// ---------------------------------------------------------------------------
// K2: color pass + H + sigmoid, in place. One wave per (b,x,y) tile.
//   a = sigmoid(gamma*(scale*(muf*Fa*P*Fa - nuf*Fb*P*Fb) + muh*h0⊗h1))
// ---------------------------------------------------------------------------
__global__ __launch_bounds__(256) void k_color(float* __restrict__ buf,
                                               const float* __restrict__ Fab,
                                               const float* __restrict__ Diff,
                                               const float* muf_p, const float* nuf_p,
                                               const float* muh_p, const float* sh_p,
                                               const float* gamma_p) {
  __shared__ float Fa[256], Fb[256];
  __shared__ float Ps[8][256], Us[8][256], Vs[8][256];
  __shared__ float H0[8][16], H1[8][16];
  int tid = threadIdx.x;
  Fa[tid] = Fab[tid];
  Fb[tid] = Fab[256 + tid];

  int wave = tid >> 5, lane = tid & 31;
  long t = (long)blockIdx.x * 8 + wave;        // grid exact: 26896*8 == 215168
  int b = (int)(t / 1681);
  int xy = (int)(t % 1681);
  float* tile = buf + t * 121;                 // 121 contiguous floats

  for (int i = lane; i < 256; i += 32) {
    int c = i >> 4, d = i & 15;
    Ps[wave][i] = (c < 11 && d < 11) ? tile[c * 11 + d] : 0.0f;
  }
  // hauss rows for H = muh * h0(c) * h1(d)
  float sh = *sh_p;
  float inv2s = 1.0f / (2.0f * sh * sh);
  if (lane < 16) {
    float v0 = 0.0f, v1 = 0.0f;
    if (lane < 11) {
      float d0 = Diff[((size_t)b * 1681 + xy) * 11 + lane];
      float d1 = Diff[((size_t)(128 + b) * 1681 + xy) * 11 + lane];
      v0 = __expf(-d0 * d0 * inv2s);
      v1 = __expf(-d1 * d1 * inv2s);
    }
    H0[wave][lane] = v0;
    H1[wave][lane] = v1;
  }
  __syncthreads();

  int n = lane & 15;
  int kb = (lane >> 4) << 1;
  int rhalf = (lane >> 4) << 3;

  // ---- Step 1: U = Fa*P, V = Fb*P ----
  v8f u = {}, v = {};
  for (int k4 = 0; k4 < 16; k4 += 4) {
    v2f af, bf, ag;
    af.x = Fa[n * 16 + k4 + kb];
    af.y = Fa[n * 16 + k4 + kb + 1];
    bf.x = Ps[wave][(k4 + kb) * 16 + n];
    bf.y = Ps[wave][(k4 + kb + 1) * 16 + n];
    u = wmma_f32(af, bf, u);
    ag.x = Fb[n * 16 + k4 + kb];
    ag.y = Fb[n * 16 + k4 + kb + 1];
    v = wmma_f32(ag, bf, v);
  }
#pragma unroll
  for (int r = 0; r < 8; ++r) {
    Us[wave][(r + rhalf) * 16 + n] = u[r];
    Vs[wave][(r + rhalf) * 16 + n] = v[r];
  }
  __syncthreads();

  // ---- Step 2: Ra = U*Fa, Rb = V*Fb ----
  v8f ra = {}, rb = {};
  for (int k4 = 0; k4 < 16; k4 += 4) {
    v2f af, bf, ag, bg;
    af.x = Us[wave][n * 16 + k4 + kb];
    af.y = Us[wave][n * 16 + k4 + kb + 1];
    bf.x = Fa[(k4 + kb) * 16 + n];
    bf.y = Fa[(k4 + kb + 1) * 16 + n];
    ra = wmma_f32(af, bf, ra);
    ag.x = Vs[wave][n * 16 + k4 + kb];
    ag.y = Vs[wave][n * 16 + k4 + kb + 1];
    bg.x = Fb[(k4 + kb) * 16 + n];
    bg.y = Fb[(k4 + kb + 1) * 16 + n];
    rb = wmma_f32(ag, bg, rb);
  }

  float muf = *muf_p, nuf = *nuf_p, muh = *muh_p, gamma = *gamma_p;
  const float scale = 1.0e-4f;                 // DX^2 * DC^2
#pragma unroll
  for (int r = 0; r < 8; ++r) {
    int c = r + rhalf;
    int d = n;
    if (c < 11 && d < 11) {
      float rv = muf * ra[r] - nuf * rb[r];
      float x = gamma * (scale * rv + muh * H0[wave][c] * H1[wave][d]);
      tile[c * 11 + d] = 1.0f / (1.0f + __expf(-x));
    }
  }
}

// ---------------------------------------------------------------------------
// K3: epilogue. Single block. Extract a[:,20,loc], build u, softmaxes, preds.
// d_out = [comp_preds(14) | a_tests(847) | as_compared(14641)]
// ---------------------------------------------------------------------------
__global__ __launch_bounds__(256) void k_final(const float* __restrict__ a,
                                               float* __restrict__ out,
                                               const int* loc_p) {
  __shared__ float uarr[847];
  int tid = threadIdx.x;
  int loc = *loc_p;
  int xy = 20 * 41 + loc;

  for (int i = tid; i < 128 * 121; i += 256) {
    int k = i / 121, j = i % 121;
    float val = a[((size_t)k * 1681 + xy) * 121 + j];
    if (k < 7) out[14 + k * 121 + j] = val;
    else out[14 + 847 + (k - 7) * 121 + j] = val;
  }

  for (int p = tid; p < 847; p += 256) {
    int nn = p / 121, k = p % 121;
    const float* tptr = a + ((size_t)nn * 1681 + xy) * 121;
    const float* cptr = a + ((size_t)(7 + k) * 1681 + xy) * 121;
    float mx = -1e30f;
    for (int j = 0; j < 121; ++j) {
      float d = fabsf(cptr[j] - tptr[j]);
      mx = fmaxf(mx, 50.0f * d);
    }
    float Z = 0.0f, S = 0.0f;
    for (int j = 0; j < 121; ++j) {
      float d = fabsf(cptr[j] - tptr[j]);
      float e = __expf(50.0f * d - mx);
      Z += e;
      S += e * d;
    }
    uarr[p] = 1.0f / (S / Z + 1.0f);
  }
  __syncthreads();

  if (tid < 7) {
    float mx = -1e30f;
    for (int k = 0; k < 121; ++k) mx = fmaxf(mx, 50.0f * uarr[tid * 121 + k]);
    float Z = 0.0f, px = 0.0f, py = 0.0f;
    for (int k = 0; k < 121; ++k) {
      float e = __expf(50.0f * uarr[tid * 121 + k] - mx);
      Z += e;
      px += e * (float)((k / 11) - 5) * 0.2f;
      py += e * (float)((k % 11) - 5) * 0.2f;
    }
    out[tid * 2 + 0] = px / Z;
    out[tid * 2 + 1] = py / Z;
  }
}

// ---------------------------------------------------------------------------
extern "C" void kernel_launch(void* const* d_in, const int* in_sizes, int n_in,
                              void* d_out, int out_size, void* d_ws, size_t ws_size,
                              hipStream_t stream) {
  (void)in_sizes; (void)n_in; (void)out_size; (void)ws_size;
  const float* Diff   = (const float*)d_in[0];
  const float* muf    = (const float*)d_in[1];
  const float* nuf    = (const float*)d_in[2];
  const float* alphaf = (const float*)d_in[3];
  const float* betaf  = (const float*)d_in[4];
  const float* mug    = (const float*)d_in[5];
  const float* nug    = (const float*)d_in[6];
  const float* alphag = (const float*)d_in[7];
  const float* betag  = (const float*)d_in[8];
  const float* muh    = (const float*)d_in[9];
  const float* sigmah = (const float*)d_in[10];
  const float* gamma  = (const float*)d_in[11];
  const int*   loc    = (const int*)d_in[12];

  float* buf0 = (float*)d_ws;
  float* buf1 = buf0 + NEL_A;
  float* Mab  = buf1 + NEL_A;        // 4608 floats
  float* Fab  = Mab + 4608;          // 512 floats

  k_mats<<<1, 256, 0, stream>>>(Mab, Fab, alphag, betag, alphaf, betaf);
  k_init<<<(int)((NEL_A + 255) / 256), 256, 0, stream>>>(buf0, NEL_A);

  float* cur = buf0;
  float* nxt = buf1;
  for (int it = 0; it < 15; ++it) {
    k_spatial<<<IMG, 128, 0, stream>>>(cur, nxt, Mab, mug, nug);
    k_color<<<NTILE / 8, 256, 0, stream>>>(nxt, Fab, Diff, muf, nuf, muh, sigmah, gamma);
    float* tmp = cur; cur = nxt; nxt = tmp;
  }
  k_final<<<1, 256, 0, stream>>>(cur, (float*)d_out, loc);
}